// RandScatterRouter_34737695490468
// MI455X (gfx1250) — compile-verified
//
#include <hip/hip_runtime.h>
#include <math.h>
#include <stdint.h>

// ---- problem constants (match reference) ----
#define N_TOKENS  16384
#define D_MODEL   2048
#define N_EXPERTS 16
#define CAPACITY  2048               // 2*N/E
#define TPB_ROUTE 256
#define NB_ROUTE  (N_TOKENS / TPB_ROUTE)   // 64 blocks
#define N_SLOTS   (N_EXPERTS * CAPACITY)   // 32768
#define ROWS_PER_BLOCK 4                   // gather rows per block (32 KB LDS)

// =====================================================================
// Kernel 0: init inverse slot map to -1 (must be re-done every call)
// =====================================================================
__global__ void k_init_slot(int* __restrict__ w_slot) {
    int i = blockIdx.x * blockDim.x + threadIdx.x;
    if (i < N_SLOTS) w_slot[i] = -1;
}

// =====================================================================
// Kernel 1: per-token argmax (first-max tiebreak, matching jnp.argmax)
//           + per-block expert histogram
// =====================================================================
__global__ void __launch_bounds__(TPB_ROUTE)
k_argmax_hist(const float* __restrict__ score,
              int*   __restrict__ w_eidx,
              int*   __restrict__ w_hist,
              float* __restrict__ out_eidx) {
    __shared__ int cnt[N_EXPERTS];
    const int tid = threadIdx.x;
    if (tid < N_EXPERTS) cnt[tid] = 0;
    __syncthreads();

    const int tok = blockIdx.x * TPB_ROUTE + tid;
    const float4* s4 = reinterpret_cast<const float4*>(score + (size_t)tok * N_EXPERTS);
    float best = -INFINITY;
    int   bi   = 0;
#pragma unroll
    for (int q = 0; q < 4; ++q) {
        float4 v = s4[q];
        float a0 = v.x, a1 = v.y, a2 = v.z, a3 = v.w;
        if (a0 > best) { best = a0; bi = q * 4 + 0; }
        if (a1 > best) { best = a1; bi = q * 4 + 1; }
        if (a2 > best) { best = a2; bi = q * 4 + 2; }
        if (a3 > best) { best = a3; bi = q * 4 + 3; }
    }
    w_eidx[tok]   = bi;
    out_eidx[tok] = (float)bi;
    atomicAdd(&cnt[bi], 1);
    __syncthreads();
    if (tid < N_EXPERTS) w_hist[blockIdx.x * N_EXPERTS + tid] = cnt[tid];
}

// =====================================================================
// Kernel 2: exclusive scan of per-block histograms across blocks
//           (64 blocks x 16 experts -> trivial; 16 working threads)
// =====================================================================
__global__ void k_scan_hist(const int* __restrict__ w_hist,
                            int* __restrict__ w_off) {
    const int e = threadIdx.x;
    if (e >= N_EXPERTS) return;
    int run = 0;
    for (int b = 0; b < NB_ROUTE; ++b) {
        w_off[b * N_EXPERTS + e] = run;
        run += w_hist[b * N_EXPERTS + e];
    }
}

// =====================================================================
// Kernel 3: within-block rank + global position, keep flag, and the
//           inverse slot map (slot -> token). Positions are globally
//           unique per expert, so slot-map writes are race-free.
// =====================================================================
__global__ void __launch_bounds__(TPB_ROUTE)
k_pos_slot(const int* __restrict__ w_eidx,
           const int* __restrict__ w_off,
           int*   __restrict__ w_slot,
           float* __restrict__ out_pos,
           float* __restrict__ out_keep) {
    __shared__ int sh_e[TPB_ROUTE];
    const int tid = threadIdx.x;
    const int tok = blockIdx.x * TPB_ROUTE + tid;
    const int e   = w_eidx[tok];
    sh_e[tid] = e;
    __syncthreads();

    int rank = 0;
    for (int j = 0; j < tid; ++j) rank += (sh_e[j] == e) ? 1 : 0;

    const int pos = w_off[blockIdx.x * N_EXPERTS + e] + rank;
    out_pos[tok]  = (float)pos;
    const bool keep = (pos < CAPACITY);
    out_keep[tok] = keep ? 1.0f : 0.0f;
    if (keep) w_slot[e * CAPACITY + pos] = tok;
}

// =====================================================================
// Kernel 4: gather. 4 (expert,slot) rows of 2048 floats per block.
// 512 threads = 16 wave32s, one b128 (16B) per lane per row.
// gfx1250: stream rows global -> LDS -> global with the async copy
// engine (ASYNCcnt). All 4 loads issued before the single wait for
// 4x the in-flight bytes per wave; zero-rows are stored directly
// while the async loads are still in flight.
// =====================================================================
#if defined(__gfx1250__) && \
    __has_builtin(__builtin_amdgcn_global_load_async_to_lds_b128) && \
    __has_builtin(__builtin_amdgcn_global_store_async_from_lds_b128) && \
    __has_builtin(__builtin_amdgcn_s_wait_asynccnt)
#define USE_ASYNC_LDS 1
#else
#define USE_ASYNC_LDS 0
#endif

#if USE_ASYNC_LDS
// Builtin parameter type per hipcc diagnostic: int vector_size(16) pointers,
// global (AS1) for memory side, LDS (AS3) for the shared side.
typedef int v4i __attribute__((__vector_size__(16)));
typedef __attribute__((address_space(1))) v4i* gptr_v4i;
typedef __attribute__((address_space(3))) v4i* lptr_v4i;
#endif

__global__ void __launch_bounds__(512)
k_gather(const float* __restrict__ inputs,
         const int*   __restrict__ w_slot,
         float*       __restrict__ disp) {
    const int tid  = threadIdx.x;                      // 0..511, 16B each
    const int base = blockIdx.x * ROWS_PER_BLOCK;      // first slot of block

    int toks[ROWS_PER_BLOCK];
#pragma unroll
    for (int r = 0; r < ROWS_PER_BLOCK; ++r)
        toks[r] = w_slot[base + r];                    // block-uniform

#if USE_ASYNC_LDS
    __shared__ v4i stage[ROWS_PER_BLOCK][512];         // 32 KB staging

    // Phase 1: fire async loads for all occupied rows (uniform branches).
#pragma unroll
    for (int r = 0; r < ROWS_PER_BLOCK; ++r) {
        if (toks[r] >= 0) {
            const float4* src =
                reinterpret_cast<const float4*>(inputs + (size_t)toks[r] * D_MODEL);
            __builtin_amdgcn_global_load_async_to_lds_b128(
                (gptr_v4i)(src + tid), (lptr_v4i)(&stage[r][tid]), 0, 0);
        }
    }
    // Overlap: empty rows are pure zero-fill, no LDS round-trip needed.
#pragma unroll
    for (int r = 0; r < ROWS_PER_BLOCK; ++r) {
        if (toks[r] < 0) {
            float4* dst = reinterpret_cast<float4*>(disp + (size_t)(base + r) * D_MODEL);
            dst[tid] = make_float4(0.f, 0.f, 0.f, 0.f);
        }
    }
    // Phase 2: drain loads once, then fire all async stores.
    __builtin_amdgcn_s_wait_asynccnt(0);
#pragma unroll
    for (int r = 0; r < ROWS_PER_BLOCK; ++r) {
        if (toks[r] >= 0) {
            float4* dst = reinterpret_cast<float4*>(disp + (size_t)(base + r) * D_MODEL);
            __builtin_amdgcn_global_store_async_from_lds_b128(
                (gptr_v4i)(dst + tid), (lptr_v4i)(&stage[r][tid]), 0, 0);
        }
    }
    __builtin_amdgcn_s_wait_asynccnt(0);
#else
#pragma unroll
    for (int r = 0; r < ROWS_PER_BLOCK; ++r) {
        float4* dst = reinterpret_cast<float4*>(disp + (size_t)(base + r) * D_MODEL);
        if (toks[r] < 0) {
            dst[tid] = make_float4(0.f, 0.f, 0.f, 0.f);
        } else {
            const float4* src =
                reinterpret_cast<const float4*>(inputs + (size_t)toks[r] * D_MODEL);
            dst[tid] = src[tid];
        }
    }
#endif
}

// =====================================================================
// launcher
// =====================================================================
extern "C" void kernel_launch(void* const* d_in, const int* in_sizes, int n_in,
                              void* d_out, int out_size, void* d_ws, size_t ws_size,
                              hipStream_t stream) {
    (void)in_sizes; (void)n_in; (void)out_size; (void)ws_size;

    const float* inputs = (const float*)d_in[0];   // [N, D] f32
    const float* score  = (const float*)d_in[1];   // [N, E] f32

    // d_out layout: dispatched [E,C,D] | expert_idx [N] | pos [N] | keep [N]
    float* out      = (float*)d_out;
    float* out_disp = out;
    float* out_eidx = out_disp + (size_t)N_SLOTS * D_MODEL;
    float* out_pos  = out_eidx + N_TOKENS;
    float* out_keep = out_pos  + N_TOKENS;

    // workspace layout (bytes): eidx | hist | off | slot_map
    char* ws     = (char*)d_ws;
    int*  w_eidx = (int*)(ws);                         // N * 4        = 65536
    int*  w_hist = (int*)(ws + 65536);                 // NB*E * 4     = 4096
    int*  w_off  = (int*)(ws + 65536 + 4096);          // NB*E * 4     = 4096
    int*  w_slot = (int*)(ws + 65536 + 4096 + 4096);   // E*C * 4      = 131072

    k_init_slot  <<<(N_SLOTS + 255) / 256, 256, 0, stream>>>(w_slot);
    k_argmax_hist<<<NB_ROUTE, TPB_ROUTE, 0, stream>>>(score, w_eidx, w_hist, out_eidx);
    k_scan_hist  <<<1, 32, 0, stream>>>(w_hist, w_off);
    k_pos_slot   <<<NB_ROUTE, TPB_ROUTE, 0, stream>>>(w_eidx, w_off, w_slot,
                                                      out_pos, out_keep);
    k_gather     <<<N_SLOTS / ROWS_PER_BLOCK, 512, 0, stream>>>(inputs, w_slot, out_disp);
}